// KDALayer_62302795596584
// MI455X (gfx1250) — compile-verified
//
#include <hip/hip_runtime.h>
#include <math.h>
#include <stdint.h>

// ---- problem sizes (compile-time) ----
#define BB 4
#define TT 2048
#define DM 1024
#define NH 16
#define DH 64
#define DI 1024
#define MROWS (BB*TT)   // 8192

#define KCHUNK 32       // K elements staged per TDM transfer
#define ASTRIDE 36      // padded LDS row stride (DWORDs): 32 + 4 pad -> bank-conflict-free

typedef __attribute__((ext_vector_type(2))) float v2f;
typedef __attribute__((ext_vector_type(8))) float v8f;
typedef __attribute__((ext_vector_type(4))) unsigned int v4u;
typedef __attribute__((ext_vector_type(4))) int v4i;
typedef __attribute__((ext_vector_type(8))) int v8i;

__device__ __forceinline__ float sigmoidf_(float x) {
    return 1.0f / (1.0f + __expf(-x));
}

// ---------------------------------------------------------------------------
// TDM: DMA a 64-row x KCHUNK-col f32 tile (row stride K elems) into LDS with
// 4-DWORD padding after every 32 DWORDs => LDS row stride ASTRIDE.
// D# layout per CDNA5 ISA 8.3/8.4 (group0 128b, group1 256b), 2D tensor so
// groups 2/3 are zero. 6-arg builtin form (clang-23 lane).
// ---------------------------------------------------------------------------
__device__ __forceinline__ void tdm_load_a_tile(const float* gptr,
                                                unsigned lds_off, int K)
{
    const unsigned long long ga = (unsigned long long)(uintptr_t)gptr;
    v4u g0;
    g0.x = 1u;                                                  // count=1, user mode
    g0.y = lds_off;                                             // lds_addr (bytes)
    g0.z = (unsigned)(ga & 0xFFFFFFFFull);                      // global_addr[31:0]
    g0.w = (unsigned)((ga >> 32) & 0x1FFFFFFull) | (2u << 30);  // ga[56:32] | type=2

    v8i g1;
    g1[0] = (int)((2u << 16)      // data_size = 2 (4 bytes)
                | (1u << 20)      // pad_enable
                | (4u << 22)      // pad_interval: 32 DWORDs
                | (3u << 25));    // pad_amount:   4 DWORDs
    g1[1] = (int)(((unsigned)K & 0xFFFFu) << 16);        // tensor_dim0[15:0]
    g1[2] = (int)((((unsigned)K >> 16) & 0xFFFFu)        // tensor_dim0[31:16]
                | ((unsigned)MROWS << 16));              // tensor_dim1[15:0]
    g1[3] = (int)(((unsigned)MROWS >> 16)                // tensor_dim1[31:16]
                | ((unsigned)KCHUNK << 16));             // tile_dim0 = 32
    g1[4] = 64;                                          // tile_dim1 = 64 rows (tile_dim2=0)
    g1[5] = K;                                           // tensor_dim0_stride[31:0]
    g1[6] = 0;                                           // stride hi / dim1_stride lo
    g1[7] = 0;

    const v4i z4 = {0, 0, 0, 0};
    const v8i z8 = {0, 0, 0, 0, 0, 0, 0, 0};
    __builtin_amdgcn_tensor_load_to_lds(g0, g1, z4, z4, z8, 0);
}

// ---------------------------------------------------------------------------
// LayerNorm: one 256-thread block per (b,t) row of 1024
// ---------------------------------------------------------------------------
__global__ void __launch_bounds__(256) ln_kernel(
    const float* __restrict__ x, const float* __restrict__ g,
    const float* __restrict__ be, float* __restrict__ xn)
{
    const int row = blockIdx.x;
    const float* xr = x + (size_t)row * DM;
    float* xo = xn + (size_t)row * DM;

    float vals[4];
    float s = 0.f, ss = 0.f;
    #pragma unroll
    for (int i = 0; i < 4; ++i) {
        float v = xr[threadIdx.x + i * 256];
        vals[i] = v; s += v; ss += v * v;
    }
    __shared__ float r0[256], r1[256];
    r0[threadIdx.x] = s; r1[threadIdx.x] = ss;
    __syncthreads();
    for (int off = 128; off > 0; off >>= 1) {
        if (threadIdx.x < off) {
            r0[threadIdx.x] += r0[threadIdx.x + off];
            r1[threadIdx.x] += r1[threadIdx.x + off];
        }
        __syncthreads();
    }
    const float mu   = r0[0] * (1.0f / DM);
    const float var  = r1[0] * (1.0f / DM) - mu * mu;
    const float rstd = rsqrtf(var + 1e-5f);
    #pragma unroll
    for (int i = 0; i < 4; ++i) {
        int c = threadIdx.x + i * 256;
        xo[c] = (vals[i] - mu) * rstd * g[c] + be[c];
    }
}

// ---------------------------------------------------------------------------
// f32 WMMA GEMM, TDM-staged A + 4x4 register blocking.
// Block = 8 waves sharing one 64-row stripe (tm); wave w owns macro-col
// tn = (blockIdx%2)*8 + w. A 64xKCHUNK chunk is DMA'd to LDS by the TDM
// (double-buffered, wave 0 issues, tensorcnt + barrier publishes), so A global
// traffic drops 8x vs per-wave loads. Each wave computes a 64x64 macro-tile
// as 4x4 grid of 16x16 V_WMMA_F32_16X16X4_F32.
// EPI: 0 = plain, 1 = sigmoid(x + bias[col]), 2 = x + resid[row,col]
// ---------------------------------------------------------------------------
template <int EPI>
__global__ void __launch_bounds__(256) gemm_wmma_f32(
    const float* __restrict__ A, const float* __restrict__ W,
    const float* __restrict__ bias, const float* __restrict__ resid,
    float* __restrict__ out, int N, int K)
{
    __shared__ float As[2][64 * ASTRIDE];     // 2 x 9 KB double buffer

    const int lane   = threadIdx.x & 31;
    const int wave   = threadIdx.x >> 5;
    const int tilesN = N >> 6;                // 64-wide macro-cols (16 here)
    const int bpr    = tilesN >> 3;           // blocks per macro-row (2)
    const int tm     = blockIdx.x / bpr;      // shared by all 8 waves
    const int tn     = (blockIdx.x - tm * bpr) * 8 + wave;

    const int mlane = lane & 15;
    const int khalf = (lane >> 4) << 1;       // 0 for lanes 0-15, 2 for 16-31
    const int col0  = tn * 64 + mlane;

    const float* atile = A + (size_t)(tm * 64) * K;   // 64-row stripe origin
    const float* bcol0 = W + col0 + (size_t)khalf * N;

    v8f acc[4][4] = {};

    // prologue: DMA chunk 0
    if (wave == 0) {
        tdm_load_a_tile(atile, (unsigned)(uintptr_t)&As[0][0], K);
    }

    const int NC = K / KCHUNK;
    for (int c = 0; c < NC; ++c) {
        if (wave == 0) {
            if (c + 1 < NC) {
                // next chunk into the buffer freed at end of iteration c-1
                tdm_load_a_tile(atile + (c + 1) * KCHUNK,
                                (unsigned)(uintptr_t)&As[(c + 1) & 1][0], K);
                __builtin_amdgcn_s_wait_tensorcnt(1);   // chunk c complete
            } else {
                __builtin_amdgcn_s_wait_tensorcnt(0);
            }
        }
        __syncthreads();                       // publish chunk c to all waves

        const float* Ab = &As[c & 1][0];
        const int kc = c * KCHUNK;

        #pragma unroll
        for (int kk = 0; kk < KCHUNK; kk += 4) {
            if (kk == 0) {
                __builtin_prefetch(bcol0 + (size_t)(kc + KCHUNK) * N, 0, 3);
            }
            // A fragments from LDS (ds_load_b64, conflict-free via ASTRIDE=36)
            v2f a[4];
            #pragma unroll
            for (int i = 0; i < 4; ++i) {
                const float2 av =
                    *(const float2*)&Ab[(mlane + 16 * i) * ASTRIDE + kk + khalf];
                a[i].x = av.x; a[i].y = av.y;
            }
            // B fragments from global: lane(0-15) W[k+0..1][col], lane(16-31) W[k+2..3][col]
            v2f b[4];
            #pragma unroll
            for (int j = 0; j < 4; ++j) {
                b[j].x = bcol0[(size_t)(kc + kk) * N + 16 * j];
                b[j].y = bcol0[(size_t)(kc + kk) * N + N + 16 * j];
            }
            #pragma unroll
            for (int i = 0; i < 4; ++i) {
                #pragma unroll
                for (int j = 0; j < 4; ++j) {
                    acc[i][j] = __builtin_amdgcn_wmma_f32_16x16x4_f32(
                        false, a[i], false, b[j], (short)0, acc[i][j], false, false);
                }
            }
        }
        __syncthreads();   // all reads of buffer (c&1) done before TDM(c+2) lands
    }

    // D layout per 16x16 tile: VGPR r -> M = r (lanes 0-15) / 8+r (lanes 16-31)
    const int rhalf = (lane < 16) ? 0 : 8;
    #pragma unroll
    for (int j = 0; j < 4; ++j) {
        const int cIdx = tn * 64 + 16 * j + mlane;
        float bc = 0.f;
        if (EPI == 1) bc = bias[cIdx];
        #pragma unroll
        for (int i = 0; i < 4; ++i) {
            #pragma unroll
            for (int r8 = 0; r8 < 8; ++r8) {
                const int r = tm * 64 + 16 * i + rhalf + r8;
                float v = acc[i][j][r8];
                if (EPI == 1)      v = sigmoidf_(v + bc);
                else if (EPI == 2) v = v + resid[(size_t)r * N + cIdx];
                out[(size_t)r * N + cIdx] = v;
            }
        }
    }
}

// ---------------------------------------------------------------------------
// beta = sigmoid(xn @ W_beta + b_beta): one thread per (row, head)
// ---------------------------------------------------------------------------
__global__ void __launch_bounds__(256) beta_kernel(
    const float* __restrict__ xn, const float* __restrict__ Wb,
    const float* __restrict__ bb, float* __restrict__ beta)
{
    const int gid = blockIdx.x * 256 + threadIdx.x;   // [0, MROWS*NH)
    const int row = gid >> 4;
    const int h   = gid & 15;
    const float* xr = xn + (size_t)row * DM;
    float s = bb[h];
    #pragma unroll 4
    for (int j = 0; j < DM; ++j) s += xr[j] * Wb[j * NH + h];
    beta[gid] = sigmoidf_(s);
}

// ---------------------------------------------------------------------------
// Gated delta-rule scan. One 64-thread block per (b,h); thread e owns S[:,e]
// in 64 VGPRs. q/k/alpha broadcast via LDS; sequential over T.
// ---------------------------------------------------------------------------
__global__ void __launch_bounds__(64) scan_kernel(
    const float* __restrict__ q, const float* __restrict__ k,
    const float* __restrict__ v, const float* __restrict__ alpha,
    const float* __restrict__ beta, float* __restrict__ o)
{
    const int bh = blockIdx.x;
    const int b  = bh >> 4;
    const int h  = bh & 15;
    const int e  = threadIdx.x;

    __shared__ float q_s[DH], k_s[DH], a_s[DH];

    float S[DH];
    #pragma unroll
    for (int d = 0; d < DH; ++d) S[d] = 0.f;

    const size_t base = ((size_t)b * TT) * DI + h * DH + e;
    const size_t bhb  = ((size_t)b * TT) * NH + h;

    for (int t = 0; t < TT; ++t) {
        const size_t idx = base + (size_t)t * DI;
        const float qx = q[idx];
        const float kx = k[idx];
        const float ax = alpha[idx];   // sigmoid already applied
        const float vx = v[idx];
        const float bt = beta[bhb + (size_t)t * NH];  // broadcast load

        __syncthreads();               // protect LDS from previous iteration
        q_s[e] = qx; k_s[e] = kx; a_s[e] = ax;
        __syncthreads();

        // ||k||: redundant per-thread reduction over LDS (broadcast reads)
        float ss = 0.f;
        #pragma unroll
        for (int d = 0; d < DH; ++d) { const float kd = k_s[d]; ss += kd * kd; }
        const float inv = 1.0f / fmaxf(sqrtf(ss), 1e-12f);

        // decay + projection:  proj_e = inv * sum_d k_d * (a_d * S[d][e])
        float proj = 0.f;
        #pragma unroll
        for (int d = 0; d < DH; ++d) {
            const float s2 = S[d] * a_s[d];
            S[d] = s2;
            proj += k_s[d] * s2;
        }
        proj *= inv;

        // rank-1 update fused with output dot: S[d][e] += bt*kn_d*(v_e - proj_e)
        const float coef = bt * (vx - proj) * inv;
        float oe = 0.f;
        #pragma unroll
        for (int d = 0; d < DH; ++d) {
            const float s2 = S[d] + coef * k_s[d];
            S[d] = s2;
            oe += q_s[d] * s2;
        }
        o[idx] = oe;
    }
}

// ---------------------------------------------------------------------------
extern "C" void kernel_launch(void* const* d_in, const int* in_sizes, int n_in,
                              void* d_out, int out_size, void* d_ws, size_t ws_size,
                              hipStream_t stream)
{
    (void)in_sizes; (void)n_in; (void)out_size; (void)ws_size;

    const float* x   = (const float*)d_in[0];
    const float* Wq  = (const float*)d_in[1];
    const float* Wk  = (const float*)d_in[2];
    const float* Wv  = (const float*)d_in[3];
    const float* Wa  = (const float*)d_in[4];
    const float* ba  = (const float*)d_in[5];
    const float* Wb  = (const float*)d_in[6];
    const float* bb  = (const float*)d_in[7];
    const float* Wo  = (const float*)d_in[8];
    const float* lg  = (const float*)d_in[9];
    const float* lb  = (const float*)d_in[10];
    float* out = (float*)d_out;

    float* ws = (float*)d_ws;
    const size_t SZ = (size_t)MROWS * DI;   // 8192*1024 floats
    float* xn   = ws;
    float* qb   = ws + 1 * SZ;
    float* kb   = ws + 2 * SZ;
    float* vb   = ws + 3 * SZ;
    float* ab   = ws + 4 * SZ;
    float* btb  = ws + 5 * SZ;              // MROWS*NH floats
    float* ob   = xn;                        // xn dead after projections -> reuse

    // 1. LayerNorm
    ln_kernel<<<MROWS, 256, 0, stream>>>(x, lg, lb, xn);

    // 2. WMMA projections, TDM-staged A, 64x64 macro-tile per wave
    const int gemm_blocks = (MROWS / 64) * (DI / 64) / 8;   // 256
    gemm_wmma_f32<0><<<gemm_blocks, 256, 0, stream>>>(xn, Wq, nullptr, nullptr, qb, DI, DM);
    gemm_wmma_f32<0><<<gemm_blocks, 256, 0, stream>>>(xn, Wk, nullptr, nullptr, kb, DI, DM);
    gemm_wmma_f32<0><<<gemm_blocks, 256, 0, stream>>>(xn, Wv, nullptr, nullptr, vb, DI, DM);
    gemm_wmma_f32<1><<<gemm_blocks, 256, 0, stream>>>(xn, Wa, ba,      nullptr, ab, DI, DM);

    // 3. beta
    beta_kernel<<<(MROWS * NH) / 256, 256, 0, stream>>>(xn, Wb, bb, btb);

    // 4. sequential scan (64 chains, register-resident state)
    scan_kernel<<<BB * NH, 64, 0, stream>>>(qb, kb, vb, ab, btb, ob);

    // 5. output projection + residual
    gemm_wmma_f32<2><<<gemm_blocks, 256, 0, stream>>>(ob, Wo, nullptr, x, out, DM, DI);
}